// PoseLoss_15006615732299
// MI455X (gfx1250) — compile-verified
//
#include <hip/hip_runtime.h>
#include <math.h>

#define B_ 16
#define N_ 1024
#define PX_THRESH2 9.0f   // PX_THRESH^2

typedef __attribute__((ext_vector_type(2))) float v2f;
typedef __attribute__((ext_vector_type(8))) float v8f;
typedef __attribute__((ext_vector_type(4))) unsigned int v4u;
typedef __attribute__((ext_vector_type(8))) int v8i;
typedef __attribute__((ext_vector_type(4))) int v4i;

// ---------------------------------------------------------------------------
// Kernel 1: nearest-neighbour argmin via V_WMMA_F32_16X16X4_F32 distance tiles
//   D[i][j] = ||q_i||^2 - 2 q_i.b_j + ||b_j||^2
//   A row (M=i): [qx, qy, ||q||^2, 1]   (16x4)
//   B col (N=j): [-2bx, -2by, 1, ||b||^2]^T  (4x16)
// Target points staged into LDS by the Tensor Data Mover (one descriptor,
// TENSORcnt-tracked), then 8 waves each own 8 row-tiles of 16 rows.
// ---------------------------------------------------------------------------
__global__ __launch_bounds__(256) void nn_argmin_kernel(
    const float* __restrict__ q_pts,   // [B][N][2] queries
    const float* __restrict__ t_pts,   // [B][N][2] targets
    int*  __restrict__ nn_idx,         // [B][N]
    float* __restrict__ nn_dsq)        // [B][N] squared distance to NN
{
    __shared__ float traw[N_ * 2];     // raw interleaved (x,y), filled by TDM
    __shared__ float tbn[N_];          // ||b||^2

    const int b    = blockIdx.x;
    const int tid  = threadIdx.x;
    const int lane = tid & 31;
    const int wave = tid >> 5;
    const int m    = lane & 15;
    const bool hi  = lane >= 16;

    // hint the query rows into cache while the TDM runs
    __builtin_prefetch(q_pts + (size_t)b * N_ * 2 + (size_t)tid * 8, 0, 1);

    if (wave == 0) {
        // ---- Tensor DMA: copy 2048 contiguous f32 (8KB) global -> LDS ----
        const unsigned long long gaddr =
            (unsigned long long)(const void*)(t_pts + (size_t)b * N_ * 2);
        const unsigned lds_off = (unsigned)(size_t)(&traw[0]); // low 32 bits = LDS offset

        v4u g0 = {0u, 0u, 0u, 0u};
        g0.x = 1u;                                  // count = 1 (valid descriptor)
        g0.y = lds_off;                             // lds_addr [63:32]
        g0.z = (unsigned)(gaddr & 0xFFFFFFFFull);   // global_addr [95:64]
        g0.w = (unsigned)((gaddr >> 32) & 0x01FFFFFFull) | (2u << 30); // [120:96] + type=2

        v8i g1 = {0, 0, 0, 0, 0, 0, 0, 0};
        g1.s0 = (int)(2u << 16);          // data_size = 2 (4 bytes); mask=0
        g1.s1 = (int)(2048u << 16);       // tensor_dim0[15:0] at bits[63:48]
        g1.s2 = (int)(1u << 16);          // tensor_dim1 = 1 (bits[111:80] low half)
        g1.s3 = (int)(2048u << 16);       // tile_dim0 = 2048 (bits[127:112])
        g1.s4 = 0;                        // tile_dim1 = tile_dim2 = 0 (unused)
        g1.s5 = 2048;                     // tensor_dim0_stride[31:0]
        g1.s6 = 0;
        g1.s7 = 0;

        v4i gz4 = {0, 0, 0, 0};           // groups 2/3 unused (2-D descriptor)
        v8i gz8 = {0, 0, 0, 0, 0, 0, 0, 0};

        __builtin_amdgcn_tensor_load_to_lds(g0, g1, gz4, gz4, gz8, 0);
        __builtin_amdgcn_s_wait_tensorcnt(0);
    }
    __syncthreads();

    for (int j = tid; j < N_; j += 256) {
        const float x = traw[2 * j], y = traw[2 * j + 1];
        tbn[j] = x * x + y * y;
    }
    __syncthreads();

    for (int rt = wave; rt < 64; rt += 8) {
        const int qrow = rt * 16 + m;
        const float qx = q_pts[(b * N_ + qrow) * 2 + 0];
        const float qy = q_pts[(b * N_ + qrow) * 2 + 1];
        // A operand: lanes 0-15 hold K=0,1 ; lanes 16-31 hold K=2,3
        v2f aop;
        aop.x = hi ? (qx * qx + qy * qy) : qx;
        aop.y = hi ? 1.0f : qy;

        float minv[8];
        int   mini[8];
#pragma unroll
        for (int r = 0; r < 8; ++r) { minv[r] = 3.0e38f; mini[r] = 0; }

        for (int ct = 0; ct < 64; ++ct) {
            const int col = ct * 16 + m;
            const float bx = traw[2 * col], by = traw[2 * col + 1];
            const float bn = tbn[col];
            // B operand: lanes 0-15 hold rows K=0,1 ; lanes 16-31 rows K=2,3
            v2f bop;
            bop.x = hi ? 1.0f : (-2.0f * bx);
            bop.y = hi ? bn   : (-2.0f * by);
            v8f c = {0.f,0.f,0.f,0.f,0.f,0.f,0.f,0.f};
            v8f d = __builtin_amdgcn_wmma_f32_16x16x4_f32(
                false, aop, false, bop, (short)0, c, false, false);
#pragma unroll
            for (int r = 0; r < 8; ++r) {
                float dv = d[r];
                // strict < keeps the earliest (smallest) column index on ties
                if (dv < minv[r]) { minv[r] = dv; mini[r] = col; }
            }
        }

        // reduce the 16 column-slots of each row across the 16-lane half-wave
#pragma unroll
        for (int r = 0; r < 8; ++r) {
            float v = minv[r];
            int   i = mini[r];
            for (int moff = 1; moff < 16; moff <<= 1) {
                float ov = __shfl_xor(v, moff, 16);
                int   oi = __shfl_xor(i, moff, 16);
                if (ov < v || (ov == v && oi < i)) { v = ov; i = oi; }
            }
            if (m == 0) {
                // C/D layout: VGPR r -> row r (lanes 0-15) / row r+8 (lanes 16-31)
                int row = rt * 16 + (hi ? r + 8 : r);
                nn_idx[b * N_ + row] = i;
                nn_dsq[b * N_ + row] = v;
            }
        }
    }
}

__device__ inline void inv3(const float* a, float* o) {
    float c00 =  a[4] * a[8] - a[5] * a[7];
    float c01 = -(a[3] * a[8] - a[5] * a[6]);
    float c02 =  a[3] * a[7] - a[4] * a[6];
    float det = a[0] * c00 + a[1] * c01 + a[2] * c02;
    float id  = 1.0f / det;
    o[0] = c00 * id; o[1] = (a[2]*a[7]-a[1]*a[8]) * id; o[2] = (a[1]*a[5]-a[2]*a[4]) * id;
    o[3] = c01 * id; o[4] = (a[0]*a[8]-a[2]*a[6]) * id; o[5] = (a[2]*a[3]-a[0]*a[5]) * id;
    o[6] = c02 * id; o[7] = (a[1]*a[6]-a[0]*a[7]) * id; o[8] = (a[0]*a[4]-a[1]*a[3]) * id;
}

// ---------------------------------------------------------------------------
// Kernel 2: per-point epipolar 9-vectors into LDS (mask-zeroed), then wave 0
// accumulates the 9x9 (padded 16x16) Gram matrix A = P^T P with 256 chained
// V_WMMA_F32_16X16X4_F32 ops. One workgroup per batch.
// ---------------------------------------------------------------------------
__global__ __launch_bounds__(256) void build_gram_kernel(
    const float* __restrict__ kp1, const float* __restrict__ kp2,
    const int* __restrict__ m1,    const int* __restrict__ m2,
    const float* __restrict__ ss1, const float* __restrict__ ss2,
    const float* __restrict__ K1,  const float* __restrict__ K2,
    const int* __restrict__ nn1,   const int* __restrict__ nn2,
    const float* __restrict__ dsq1,
    float* __restrict__ Aout,      // [B][16][16]
    int*   __restrict__ cntOut)    // [B]
{
    __shared__ float Pa[N_][16];
    __shared__ int   s_cnt;

    const int b   = blockIdx.x;
    const int tid = threadIdx.x;
    if (tid == 0) s_cnt = 0;
    __syncthreads();

    float k1[9], k2[9], ik1[9], ik2[9], T[9];
    for (int i = 0; i < 9; ++i) { k1[i] = K1[b*9+i]; k2[i] = K2[b*9+i]; }
    inv3(k1, ik1);
    inv3(k2, ik2);
    for (int i = 0; i < 3; ++i)
        for (int j = 0; j < 3; ++j)
            T[i*3+j] = k1[i*3+0]*ik2[0*3+j] + k1[i*3+1]*ik2[1*3+j] + k1[i*3+2]*ik2[2*3+j];

    const float sh1x = ss1[b*4+0], sh1y = ss1[b*4+1], sc1x = ss1[b*4+2], sc1y = ss1[b*4+3];
    const float sh2x = ss2[b*4+0], sh2y = ss2[b*4+1], sc2x = ss2[b*4+2], sc2y = ss2[b*4+3];

    int local_cnt = 0;
    for (int i = tid; i < N_; i += 256) {
        const int nn = nn1[b*N_ + i];
        const bool mut = (nn2[b*N_ + nn] == i) && (dsq1[b*N_ + i] <= PX_THRESH2) &&
                         (m1[b*N_ + i] != 0) && (m2[b*N_ + nn] != 0);
        const float r1x = kp1[(b*N_+i)*2+0]  * sc1x + sh1x;
        const float r1y = kp1[(b*N_+i)*2+1]  * sc1y + sh1y;
        const float r2x = kp2[(b*N_+nn)*2+0] * sc2x + sh2x;
        const float r2y = kp2[(b*N_+nn)*2+1] * sc2y + sh2y;
        // remap kp2 into intrinsics1 pixel frame
        const float px = T[0]*r2x + T[1]*r2y + T[2];
        const float py = T[3]*r2x + T[4]*r2y + T[5];
        const float pz = T[6]*r2x + T[7]*r2y + T[8];
        const float x2 = px / pz, y2 = py / pz;
        // normalize both by inv(K1)
        float n1x = ik1[0]*r1x + ik1[1]*r1y + ik1[2];
        float n1y = ik1[3]*r1x + ik1[4]*r1y + ik1[5];
        float n1z = ik1[6]*r1x + ik1[7]*r1y + ik1[8];
        const float u1 = n1x / n1z, v1 = n1y / n1z;
        float n2x = ik1[0]*x2 + ik1[1]*y2 + ik1[2];
        float n2y = ik1[3]*x2 + ik1[4]*y2 + ik1[5];
        float n2z = ik1[6]*x2 + ik1[7]*y2 + ik1[8];
        const float u2 = n2x / n2z, v2 = n2y / n2z;

        const float w = mut ? 1.0f : 0.0f;   // w^2 == w for {0,1}
        local_cnt += mut ? 1 : 0;
        Pa[i][0] = w * (u2 * u1);
        Pa[i][1] = w * (u2 * v1);
        Pa[i][2] = w * u2;
        Pa[i][3] = w * (v2 * u1);
        Pa[i][4] = w * (v2 * v1);
        Pa[i][5] = w * v2;
        Pa[i][6] = w * u1;
        Pa[i][7] = w * v1;
        Pa[i][8] = w;
        Pa[i][9] = 0.f; Pa[i][10] = 0.f; Pa[i][11] = 0.f;
        Pa[i][12] = 0.f; Pa[i][13] = 0.f; Pa[i][14] = 0.f; Pa[i][15] = 0.f;
    }
    atomicAdd(&s_cnt, local_cnt);
    __syncthreads();
    if (tid == 0) cntOut[b] = s_cnt;

    if (tid < 32) {
        const int lane = tid;
        const int m    = lane & 15;
        const int koff = (lane < 16) ? 0 : 2;
        v8f acc = {0.f,0.f,0.f,0.f,0.f,0.f,0.f,0.f};
        for (int n0 = 0; n0 < N_; n0 += 4) {
            // A operand: M=m, K=koff..koff+1 -> P[n0+k][m]
            // B operand (symmetric product, mirrored layout): identical per-lane values
            v2f op;
            op.x = Pa[n0 + koff][m];
            op.y = Pa[n0 + koff + 1][m];
            acc = __builtin_amdgcn_wmma_f32_16x16x4_f32(
                false, op, false, op, (short)0, acc, false, false);
        }
#pragma unroll
        for (int r = 0; r < 8; ++r) {
            const int row = (lane < 16) ? r : r + 8;
            Aout[b * 256 + row * 16 + m] = acc[r];
        }
    }
}

// ---------------------------------------------------------------------------
// Kernel 3: per-batch 9x9 cyclic-Jacobi eigensolve (smallest eigenvector),
// GT essential matrix from extrinsics, masked-mean loss. 16 batches -> 16
// threads of one wave; thread 0 writes the scalar.
// ---------------------------------------------------------------------------
__global__ __launch_bounds__(32) void finalize_kernel(
    const float* __restrict__ Amat, const int* __restrict__ cnt,
    const float* __restrict__ ext1, const float* __restrict__ ext2,
    float* __restrict__ out)
{
    __shared__ float s_per[16], s_suc[16];
    const int t = threadIdx.x;

    if (t < 16) {
        float A[81], V[81];
        for (int i = 0; i < 9; ++i)
            for (int j = 0; j < 9; ++j) {
                A[i*9+j] = Amat[t * 256 + i * 16 + j];
                V[i*9+j] = (i == j) ? 1.0f : 0.0f;
            }
#pragma unroll 1
        for (int sweep = 0; sweep < 30; ++sweep) {
            for (int p = 0; p < 8; ++p)
                for (int q = p + 1; q < 9; ++q) {
                    const float apq = A[p*9+q];
                    if (fabsf(apq) < 1e-30f) continue;
                    const float app = A[p*9+p], aqq = A[q*9+q];
                    const float tau = (aqq - app) / (2.0f * apq);
                    const float tt  = (tau >= 0.f ? 1.f : -1.f) /
                                      (fabsf(tau) + sqrtf(1.f + tau * tau));
                    const float c = 1.f / sqrtf(1.f + tt * tt);
                    const float s = tt * c;
                    for (int k = 0; k < 9; ++k) {       // A = A*G
                        const float akp = A[k*9+p], akq = A[k*9+q];
                        A[k*9+p] = c * akp - s * akq;
                        A[k*9+q] = s * akp + c * akq;
                    }
                    for (int k = 0; k < 9; ++k) {       // A = G^T*A
                        const float apk = A[p*9+k], aqk = A[q*9+k];
                        A[p*9+k] = c * apk - s * aqk;
                        A[q*9+k] = s * apk + c * aqk;
                    }
                    for (int k = 0; k < 9; ++k) {       // V = V*G
                        const float vkp = V[k*9+p], vkq = V[k*9+q];
                        V[k*9+p] = c * vkp - s * vkq;
                        V[k*9+q] = s * vkp + c * vkq;
                    }
                }
        }
        int qmin = 0; float dmin = A[0];
        for (int k = 1; k < 9; ++k) {
            const float d = A[k*9+k];
            if (d < dmin) { dmin = d; qmin = k; }
        }
        float e[9], nrm = 0.f;
        for (int k = 0; k < 9; ++k) { e[k] = V[k*9+qmin]; nrm += e[k]*e[k]; }
        nrm = sqrtf(nrm);
        if (nrm < 1e-20f) nrm = 1e-20f;
        const float esgn = (e[8] < 0.f) ? -1.f : 1.f;
        for (int k = 0; k < 9; ++k) e[k] = e[k] * esgn / nrm;

        // GT essential matrix: rel = ext2 @ inv(ext1) (rigid inverse)
        float R1[9], R2[9], t1[3], t2[3];
        for (int r = 0; r < 3; ++r) {
            for (int c = 0; c < 3; ++c) {
                R1[r*3+c] = ext1[t*16 + r*4 + c];
                R2[r*3+c] = ext2[t*16 + r*4 + c];
            }
            t1[r] = ext1[t*16 + r*4 + 3];
            t2[r] = ext2[t*16 + r*4 + 3];
        }
        float Rr[9], tr[3];
        for (int i = 0; i < 3; ++i)
            for (int j = 0; j < 3; ++j)
                Rr[i*3+j] = R2[i*3+0]*R1[j*3+0] + R2[i*3+1]*R1[j*3+1] + R2[i*3+2]*R1[j*3+2];
        for (int i = 0; i < 3; ++i)
            tr[i] = t2[i] - (Rr[i*3+0]*t1[0] + Rr[i*3+1]*t1[1] + Rr[i*3+2]*t1[2]);
        const float S[9] = { 0.f, -tr[2],  tr[1],
                             tr[2], 0.f,  -tr[0],
                            -tr[1], tr[0], 0.f };
        float E[9];
        for (int i = 0; i < 3; ++i)
            for (int j = 0; j < 3; ++j)
                E[i*3+j] = S[i*3+0]*Rr[0*3+j] + S[i*3+1]*Rr[1*3+j] + S[i*3+2]*Rr[2*3+j];
        float gn = 0.f;
        for (int k = 0; k < 9; ++k) gn += E[k]*E[k];
        gn = sqrtf(gn);
        if (gn < 1e-8f) gn = 1e-8f;
        const float gsgn = (E[8] < 0.f) ? -1.f : 1.f;

        float per = 0.f;
        for (int k = 0; k < 9; ++k) {
            const float diff = e[k] - E[k] * gsgn / gn;
            per += diff * diff;
        }
        per = sqrtf(per);
        const float suc = (cnt[t] >= 8) ? 1.0f : 0.0f;
        s_per[t] = per * suc;
        s_suc[t] = suc;
    }
    __syncthreads();
    if (t == 0) {
        float sp = 0.f, ss = 0.f;
        for (int k = 0; k < 16; ++k) { sp += s_per[k]; ss += s_suc[k]; }
        if (ss < 1e-8f) ss = 1e-8f;
        out[0] = sp / ss;     // LOSS_LAMBDA == 1
    }
}

extern "C" void kernel_launch(void* const* d_in, const int* in_sizes, int n_in,
                              void* d_out, int out_size, void* d_ws, size_t ws_size,
                              hipStream_t stream) {
    (void)in_sizes; (void)n_in; (void)out_size; (void)ws_size;
    const float* kp1  = (const float*)d_in[0];
    const float* kp2  = (const float*)d_in[1];
    const float* wkp1 = (const float*)d_in[2];
    const float* wkp2 = (const float*)d_in[3];
    const int*   m1   = (const int*)d_in[4];
    const int*   m2   = (const int*)d_in[5];
    const float* ss1  = (const float*)d_in[6];
    const float* ss2  = (const float*)d_in[7];
    const float* K1   = (const float*)d_in[8];
    const float* K2   = (const float*)d_in[9];
    const float* e1   = (const float*)d_in[10];
    const float* e2   = (const float*)d_in[11];

    char* ws = (char*)d_ws;
    int*   nn1  = (int*)ws;   ws += (size_t)B_ * N_ * sizeof(int);
    int*   nn2  = (int*)ws;   ws += (size_t)B_ * N_ * sizeof(int);
    float* dsq1 = (float*)ws; ws += (size_t)B_ * N_ * sizeof(float);
    float* dsq2 = (float*)ws; ws += (size_t)B_ * N_ * sizeof(float);
    float* Amat = (float*)ws; ws += (size_t)B_ * 256 * sizeof(float);
    int*   cnt  = (int*)ws;   ws += (size_t)B_ * sizeof(int);

    nn_argmin_kernel<<<B_, 256, 0, stream>>>(wkp1, kp2, nn1, dsq1);
    nn_argmin_kernel<<<B_, 256, 0, stream>>>(wkp2, kp1, nn2, dsq2);
    build_gram_kernel<<<B_, 256, 0, stream>>>(kp1, kp2, m1, m2, ss1, ss2,
                                              K1, K2, nn1, nn2, dsq1, Amat, cnt);
    finalize_kernel<<<1, 32, 0, stream>>>(Amat, cnt, e1, e2, (float*)d_out);
}